// SAGE_52544629899903
// MI455X (gfx1250) — compile-verified
//
#include <hip/hip_runtime.h>

// GraphSAGE 2-layer forward for MI455X (gfx1250, wave32).
// Aggregation is L2-resident atomic scatter; dense layers use fp32 WMMA
// (V_WMMA_F32_16X16X4_F32) so results stay exact fp32.

typedef float v2f __attribute__((ext_vector_type(2)));
typedef float v8f __attribute__((ext_vector_type(8)));

#define IN_C  128
#define HID_C 128
#define OUT_C 64

// ---------------------------------------------------------------------------
// Scatter: agg[dst] += w * feat[src]  (+ degree count on first pass)
// One wave32 per edge, 4 channels per lane (C must be 128).
// ---------------------------------------------------------------------------
__global__ void sage_scatter(const float* __restrict__ feat,   // [N, C]
                             const int*   __restrict__ srcs,   // [E]
                             const int*   __restrict__ dsts,   // [E]
                             const float* __restrict__ ew,     // [E]
                             float*       __restrict__ agg,    // [N, C]
                             float*       __restrict__ deg,    // [N] or nullptr
                             int E, int C)
{
    const int lane = threadIdx.x & 31;
    const int wid  = threadIdx.x >> 5;
    const int e    = blockIdx.x * (blockDim.x >> 5) + wid;
    if (e >= E) return;

    const int   s = srcs[e];
    const int   d = dsts[e];
    const float w = ew[e];

    const int c0 = lane * (C >> 5);                       // 4 channels/lane
    const float4 xv = *(const float4*)(feat + (size_t)s * C + c0);
    float* ap = agg + (size_t)d * C + c0;
    unsafeAtomicAdd(ap + 0, xv.x * w);
    unsafeAtomicAdd(ap + 1, xv.y * w);
    unsafeAtomicAdd(ap + 2, xv.z * w);
    unsafeAtomicAdd(ap + 3, xv.w * w);

    if (deg != nullptr && lane == 0)
        unsafeAtomicAdd(deg + d, 1.0f);
}

// ---------------------------------------------------------------------------
// agg[i, :] /= max(deg[i], 1)
// ---------------------------------------------------------------------------
__global__ void sage_normalize(float* __restrict__ agg,
                               const float* __restrict__ deg,
                               int total, int C)
{
    const int i = blockIdx.x * blockDim.x + threadIdx.x;
    if (i >= total) return;
    const float dv = deg[i / C];
    agg[i] *= (dv > 1.0f) ? (1.0f / dv) : 1.0f;
}

// ---------------------------------------------------------------------------
// out = [relu]( A1 @ W1^T + A2 @ W2^T + bias )
//   A1,A2 : [N, K] row-major     W1,W2 : [M, K] row-major     out : [N, M]
//
// One wave32 computes one 16x16 tile using V_WMMA_F32_16X16X4_F32,
// accumulating over K in steps of 4.  blockDim.x = 128 (4 waves -> 64 cols).
//
// fp32 A-fragment layout (ISA 7.12.2): lanes 0-15 hold M=lane, K=k0/k0+1 in
// v[0]/v[1]; lanes 16-31 hold K=k0+2/k0+3  -> per-lane float2 load at
// k0 + 2*(lane>>4).  B (4x16) mirrors this with N=lane&15 rows of W.
// C/D: VGPR r holds (M = r + 8*(lane>>4), N = lane&15).
// ---------------------------------------------------------------------------
__global__ void sage_gemm_wmma(const float* __restrict__ A1,
                               const float* __restrict__ A2,
                               const float* __restrict__ W1,
                               const float* __restrict__ W2,
                               const float* __restrict__ bias,
                               float*       __restrict__ out,
                               int N, int K, int M, int do_relu)
{
    const int wave = threadIdx.x >> 5;
    const int lane = threadIdx.x & 31;
    const int row0 = blockIdx.x * 16;
    const int col0 = (blockIdx.y * 4 + wave) * 16;
    if (col0 >= M || row0 >= N) return;          // uniform per wave

    const int half = lane >> 4;                  // 0 | 1
    const int l15  = lane & 15;

    const float* a1row = A1 + (size_t)(row0 + l15) * K;
    const float* a2row = A2 + (size_t)(row0 + l15) * K;
    const float* w1row = W1 + (size_t)(col0 + l15) * K;
    const float* w2row = W2 + (size_t)(col0 + l15) * K;

    const float bj = bias[col0 + l15];           // bias per output column N
    union { v8f v; float f[8]; } acc;
    acc.v = (v8f){bj, bj, bj, bj, bj, bj, bj, bj};

    for (int k0 = 0; k0 < K; k0 += 4) {
        const int koff = k0 + 2 * half;
        v2f a1 = *(const v2f*)(a1row + koff);
        v2f b1 = *(const v2f*)(w1row + koff);
        acc.v = __builtin_amdgcn_wmma_f32_16x16x4_f32(
            false, a1, false, b1, (short)0, acc.v, false, false);
        v2f a2 = *(const v2f*)(a2row + koff);
        v2f b2 = *(const v2f*)(w2row + koff);
        acc.v = __builtin_amdgcn_wmma_f32_16x16x4_f32(
            false, a2, false, b2, (short)0, acc.v, false, false);
    }

#pragma unroll
    for (int r = 0; r < 8; ++r) {
        float v = acc.f[r];
        if (do_relu) v = fmaxf(v, 0.0f);
        out[(size_t)(row0 + r + 8 * half) * M + col0 + l15] = v;
    }
}

// ---------------------------------------------------------------------------
extern "C" void kernel_launch(void* const* d_in, const int* in_sizes, int n_in,
                              void* d_out, int out_size, void* d_ws, size_t ws_size,
                              hipStream_t stream)
{
    const float* x   = (const float*)d_in[0];   // [N, 128]
    const int*   ei  = (const int*)  d_in[1];   // [2, E]
    const float* ew  = (const float*)d_in[2];   // [E]
    /* d_in[3] node_type: unused by reference */
    const float* Wl1 = (const float*)d_in[4];   // [128, 128]
    const float* bl1 = (const float*)d_in[5];   // [128]
    const float* Wr1 = (const float*)d_in[6];   // [128, 128]
    const float* Wl2 = (const float*)d_in[7];   // [64, 128]
    const float* bl2 = (const float*)d_in[8];   // [64]
    const float* Wr2 = (const float*)d_in[9];   // [64, 128]

    const int N = in_sizes[0] / IN_C;           // 50000 (multiple of 16)
    const int E = in_sizes[2];                  // 800000
    const int* srcs = ei;
    const int* dsts = ei + E;

    // Workspace: agg [N,128] | h [N,128] | deg [N]
    char* ws = (char*)d_ws;
    const size_t aggBytes = (size_t)N * HID_C * sizeof(float);
    float* agg = (float*)(ws);
    float* h   = (float*)(ws + aggBytes);
    float* deg = (float*)(ws + 2 * aggBytes);

    // ---- Layer 1 -----------------------------------------------------
    hipMemsetAsync(agg, 0, aggBytes, stream);
    hipMemsetAsync(deg, 0, (size_t)N * sizeof(float), stream);

    const int edgesPerBlock = 8;                // 256 threads = 8 waves
    dim3 sblk(256);
    dim3 sgrd((E + edgesPerBlock - 1) / edgesPerBlock);
    sage_scatter<<<sgrd, sblk, 0, stream>>>(x, srcs, dsts, ew, agg, deg, E, IN_C);

    const int tot = N * HID_C;
    sage_normalize<<<(tot + 255) / 256, 256, 0, stream>>>(agg, deg, tot, HID_C);

    dim3 gblk(128);                             // 4 waves -> 64 output cols
    dim3 ggrd1(N / 16, HID_C / 64);
    sage_gemm_wmma<<<ggrd1, gblk, 0, stream>>>(agg, x, Wl1, Wr1, bl1, h,
                                               N, IN_C, HID_C, /*relu=*/1);

    // ---- Layer 2 (reuse agg buffer and degree) -----------------------
    hipMemsetAsync(agg, 0, aggBytes, stream);
    sage_scatter<<<sgrd, sblk, 0, stream>>>(h, srcs, dsts, ew, agg, nullptr, E, HID_C);
    sage_normalize<<<(tot + 255) / 256, 256, 0, stream>>>(agg, deg, tot, HID_C);

    dim3 ggrd2(N / 16, 1);                      // OUT_C = 64 -> one block-col
    sage_gemm_wmma<<<ggrd2, gblk, 0, stream>>>(agg, h, Wl2, Wr2, bl2,
                                               (float*)d_out,
                                               N, HID_C, OUT_C, /*relu=*/0);
}